// SETLayer_53515292508836
// MI455X (gfx1250) — compile-verified
//
#include <hip/hip_runtime.h>
#include <stdint.h>

#define IN_F  4096
#define OUT_F 4096
#define BATCH 4096
#define NP    90112
#define ROWS  8          // batch rows staged per block: 8 * 4096 * 4B = 128 KB LDS
                         // (2 blocks/WGP -> 256 KB of 320 KB LDS, 32 waves/WGP)

typedef __attribute__((ext_vector_type(4))) unsigned int v4u;
typedef __attribute__((ext_vector_type(8))) int          v8i;
typedef __attribute__((ext_vector_type(4))) int          v4i;

// ---------------- CSR build (per call; deterministic work) ----------------

__global__ void set_zero_k(int* __restrict__ p, int n) {
  int i = blockIdx.x * blockDim.x + threadIdx.x;
  if (i < n) p[i] = 0;
}

__global__ void set_hist_k(const int* __restrict__ conn_out, int* __restrict__ cnt) {
  int p = blockIdx.x * blockDim.x + threadIdx.x;
  if (p < NP) atomicAdd(&cnt[conn_out[p]], 1);
}

// One block of 1024 threads scans 4096 bucket counts -> exclusive offsets.
__global__ __launch_bounds__(1024) void set_scan_k(const int* __restrict__ cnt,
                                                   int* __restrict__ start,
                                                   int* __restrict__ cursor) {
  __shared__ int part[1024];
  const int t = threadIdx.x;
  const int base = t * 4;
  int c0 = cnt[base], c1 = cnt[base + 1], c2 = cnt[base + 2], c3 = cnt[base + 3];
  int s = c0 + c1 + c2 + c3;
  part[t] = s;
  __syncthreads();
  for (int off = 1; off < 1024; off <<= 1) {     // Hillis-Steele inclusive scan
    int v = (t >= off) ? part[t - off] : 0;
    __syncthreads();
    part[t] += v;
    __syncthreads();
  }
  int e0 = part[t] - s;                           // exclusive block prefix
  int e1 = e0 + c0, e2 = e1 + c1, e3 = e2 + c2;
  start[base] = e0;  start[base + 1] = e1;  start[base + 2] = e2;  start[base + 3] = e3;
  cursor[base] = e0; cursor[base + 1] = e1; cursor[base + 2] = e2; cursor[base + 3] = e3;
  if (t == 1023) start[4096] = part[1023];        // total == NP
}

__global__ void set_scatter_k(const int* __restrict__ conn_out,
                              const int* __restrict__ conn_in,
                              const float* __restrict__ weight,
                              int* __restrict__ cursor,
                              uint2* __restrict__ entries) {
  int p = blockIdx.x * blockDim.x + threadIdx.x;
  if (p < NP) {
    int j = conn_out[p];
    int pos = atomicAdd(&cursor[j], 1);
    entries[pos] = make_uint2((unsigned)conn_in[p], __float_as_uint(weight[p]));
  }
}

// ---------------- Main kernel: TDM-staged LDS gather ----------------

__global__ __launch_bounds__(512) void set_fwd_k(const float* __restrict__ x,
                                                 const float* __restrict__ bias,
                                                 const int* __restrict__ start,
                                                 const uint2* __restrict__ entries,
                                                 float* __restrict__ out) {
  __shared__ float xs[ROWS * IN_F];               // 128 KB: 8 full batch rows
  const int b0 = blockIdx.x * ROWS;

  // Wave 0 issues one TDM descriptor: 2D tile, 8 rows x 4096 f32, row-major.
  if (threadIdx.x < 32) {
    uint32_t lds_off = (uint32_t)(uintptr_t)(void*)&xs[0];  // generic addr low32 == LDS offset
    uint64_t ga = (uint64_t)(uintptr_t)x + (uint64_t)b0 * IN_F * sizeof(float);
    v4u g0;
    g0[0] = 1u;                                   // count=1 valid, is_restore=0, gather off
    g0[1] = lds_off;                              // lds_addr (bytes)
    g0[2] = (uint32_t)ga;                         // global_addr[31:0]
    g0[3] = (uint32_t)(ga >> 32) | (2u << 30);    // global_addr[56:32] | type=2 ("image")
    v8i g1;
    g1[0] = (2 << 16);                            // wg_mask=0, data_size=2 (4B), no pad/iter
    g1[1] = (int)((IN_F & 0xffff) << 16);         // tensor_dim0[15:0]=4096 at bits[63:48]
    g1[2] = (int)(((IN_F >> 16) & 0xffff) |       // tensor_dim0[31:16]
                  ((BATCH & 0xffff) << 16));      // tensor_dim1[15:0]
    g1[3] = (int)(((BATCH >> 16) & 0xffff) |      // tensor_dim1[31:16]
                  ((IN_F & 0xffff) << 16));       // tile_dim0 = 4096
    g1[4] = ROWS;                                 // tile_dim1 = 8, tile_dim2 = 0
    g1[5] = IN_F;                                 // tensor_dim0_stride[31:0] = 4096
    g1[6] = 0;                                    // stride hi / dim1_stride lo
    g1[7] = 0;
    v4i g2 = {0, 0, 0, 0};                        // 2D: groups 2/3 unused
    v4i g3 = {0, 0, 0, 0};
    v8i g4 = {0, 0, 0, 0, 0, 0, 0, 0};            // clang-23 6-arg form: extra group, zeroed
    __builtin_amdgcn_tensor_load_to_lds(g0, g1, g2, g3, g4, 0);
    __builtin_amdgcn_s_wait_tensorcnt(0);
  }
  __syncthreads();

  const int t = threadIdx.x;
#pragma unroll
  for (int k = 0; k < OUT_F / 512; ++k) {
    const int j = t + 512 * k;                    // output column; coalesced stores
    const int e0 = start[j];
    const int e1 = start[j + 1];
    float a0 = 0.f, a1 = 0.f, a2 = 0.f, a3 = 0.f;
    float a4 = 0.f, a5 = 0.f, a6 = 0.f, a7 = 0.f;
    for (int e = e0; e < e1; ++e) {
      const uint2 en = entries[e];                // one b64 load feeds 8 MACs
      const float w = __uint_as_float(en.y);
      const int ci = (int)en.x;
      a0 = fmaf(xs[ci],              w, a0);      // LDS gathers (ds_load_b32)
      a1 = fmaf(xs[IN_F + ci],       w, a1);
      a2 = fmaf(xs[2 * IN_F + ci],   w, a2);
      a3 = fmaf(xs[3 * IN_F + ci],   w, a3);
      a4 = fmaf(xs[4 * IN_F + ci],   w, a4);
      a5 = fmaf(xs[5 * IN_F + ci],   w, a5);
      a6 = fmaf(xs[6 * IN_F + ci],   w, a6);
      a7 = fmaf(xs[7 * IN_F + ci],   w, a7);
    }
    const float bz = bias[j];
    const size_t o = (size_t)b0 * OUT_F + j;
    __builtin_nontemporal_store(a0 + bz, &out[o]);                // NT: keep L2 for x/CSR
    __builtin_nontemporal_store(a1 + bz, &out[o + OUT_F]);
    __builtin_nontemporal_store(a2 + bz, &out[o + 2 * OUT_F]);
    __builtin_nontemporal_store(a3 + bz, &out[o + 3 * OUT_F]);
    __builtin_nontemporal_store(a4 + bz, &out[o + 4 * OUT_F]);
    __builtin_nontemporal_store(a5 + bz, &out[o + 5 * OUT_F]);
    __builtin_nontemporal_store(a6 + bz, &out[o + 6 * OUT_F]);
    __builtin_nontemporal_store(a7 + bz, &out[o + 7 * OUT_F]);
  }
}

// ---------------- Host-side launch ----------------

extern "C" void kernel_launch(void* const* d_in, const int* in_sizes, int n_in,
                              void* d_out, int out_size, void* d_ws, size_t ws_size,
                              hipStream_t stream) {
  const float* x        = (const float*)d_in[0];
  const float* weight   = (const float*)d_in[1];
  const float* bias     = (const float*)d_in[2];
  const int*   conn_in  = (const int*)d_in[3];
  const int*   conn_out = (const int*)d_in[4];
  float*       out      = (float*)d_out;

  // Workspace layout (~768 KB total)
  char*  ws      = (char*)d_ws;
  int*   cnt     = (int*)(ws);               // 4096 ints
  int*   start   = (int*)(ws + 16384);       // 4097 ints
  int*   cursor  = (int*)(ws + 49152);       // 4096 ints
  uint2* entries = (uint2*)(ws + 65536);     // 90112 x 8 B

  set_zero_k<<<16, 256, 0, stream>>>(cnt, OUT_F);
  set_hist_k<<<(NP + 255) / 256, 256, 0, stream>>>(conn_out, cnt);
  set_scan_k<<<1, 1024, 0, stream>>>(cnt, start, cursor);
  set_scatter_k<<<(NP + 255) / 256, 256, 0, stream>>>(conn_out, conn_in, weight, cursor, entries);
  set_fwd_k<<<BATCH / ROWS, 512, 0, stream>>>(x, bias, start, entries, out);

  (void)in_sizes; (void)n_in; (void)out_size; (void)ws_size;
}